// TorsionalEmbedding_30408368456388
// MI455X (gfx1250) — compile-verified
//
#include <hip/hip_runtime.h>
#include <math.h>

// ---------------------------------------------------------------------------
// TorsionalEmbedding for gfx1250 (MI455X).
// out[t, (a*7+b)*6 + r] = cbf[t, a*7+b] * rbf[idx_kj[t], b*6+r]
// T = 400k rows x 294 f32 = ~470MB stores -> store-BW bound (~20us @ 23.3TB/s).
// Strategy:
//   K0: Newton-refine 42 spherical Bessel zeros + norms (double) -> device globals
//   K1: RBF table (E x 48-padded) in workspace (19MB, L2-resident)
//   K2: per-lane cbf into LDS + ASYNC global->LDS b128 gather of rbf rows,
//       then wave-transposed coalesced float2 row stores.
// ---------------------------------------------------------------------------

#define NS 7
#define NK 6
#define NCBF 49      // NS*NS
#define NRBF 42      // NS*NK
#define RBF_PAD 48   // padded row (192B, 16B aligned) for b128 async gathers
#define OUT_ROW 294  // NCBF*NK

// ---------------- compile-time prefactor tables (double math, f32 cast) ----
constexpr double kPI = 3.14159265358979323846;
constexpr double cfact(int n) { double r = 1.0; for (int i = 2; i <= n; ++i) r *= i; return r; }
constexpr double csqrt(double x) {
  double g = x > 1.0 ? x : 1.0;
  for (int i = 0; i < 64; ++i) g = 0.5 * (g + x / g);
  return g;
}
struct Pref {
  float p0[NS];      // pref(l,0)
  float ps[NS][NS];  // sqrt(2)*pref(l,m)  (double product, single f32 round)
};
constexpr Pref makePref() {
  Pref t{};
  for (int l = 0; l < NS; ++l) {
    t.p0[l] = (float)csqrt((2 * l + 1) * cfact(l) / (4.0 * kPI * cfact(l)));
    for (int m = 0; m <= l; ++m) {
      double pr = csqrt((2 * l + 1) * cfact(l - m) / (4.0 * kPI * cfact(l + m)));
      t.ps[l][m] = (float)(csqrt(2.0) * pr);
    }
  }
  return t;
}
constexpr Pref PREF = makePref();

// ---------------- device state: Bessel zeros and norms ---------------------
__device__ float g_zeros[NRBF];
__device__ float g_norms[NRBF];

// Seeds accurate to ~1e-3; Newton (double) polishes to machine precision.
__constant__ double SEEDS[NRBF] = {
  3.141592653589793, 6.283185307179586, 9.424777960769380, 12.566370614359172, 15.707963267948966, 18.849555921538759,
  4.493409,          7.725252,          10.904122,          14.066194,          17.220755,          20.371303,
  5.763459,          9.095011,          12.322941,          15.514603,          18.689036,          21.853874,
  6.987932,         10.417119,          13.698023,          16.923621,          20.121806,          23.304247,
  8.182561,         11.704907,          15.039665,          18.301256,          21.525418,          24.727566,
  9.355812,         12.966530,          16.354710,          19.653152,          22.904551,          26.127750,
 10.512835,         14.207392,          17.647975,          20.983463,          24.262768,          27.507868
};

__global__ void zeros_kernel() {
  int t = threadIdx.x;
  if (t >= NRBF) return;
  int l = t / NK;
  double z = SEEDS[t];
  for (int it = 0; it < 12; ++it) {
    double s = sin(z), c = cos(z);
    double jm = c / z;  // j_{-1}
    double j  = s / z;  // j_0
    for (int i = 0; i < l; ++i) { double nj = (2 * i + 1) / z * j - jm; jm = j; j = nj; }
    double fp = jm - (double)(l + 1) / z * j;  // j_l'
    z -= j / fp;
  }
  // norm = 1/sqrt(0.5*j_{l+1}(z)^2)
  double s = sin(z), c = cos(z);
  double jm = c / z, j = s / z;
  for (int i = 0; i < l + 1; ++i) { double nj = (2 * i + 1) / z * j - jm; jm = j; j = nj; }
  g_zeros[t] = (float)z;
  g_norms[t] = (float)(1.0 / sqrt(0.5 * j * j));
}

// ---------------- RBF row (mirrors reference f32 op order) -----------------
__device__ inline void compute_rbf_row(float x, float* __restrict__ dst) {
#pragma unroll
  for (int l = 0; l < NS; ++l) {
#pragma unroll
    for (int k = 0; k < NK; ++k) {
      float arg = x * g_zeros[l * NK + k];
      float s, c;
      sincosf(arg, &s, &c);
      float j = s / arg;
      float res;
      if (l == 0) {
        res = j;
      } else {
        float jn = s / (arg * arg) - c / arg;
#pragma unroll
        for (int i = 1; i < l; ++i) { float nx = (float)(2 * i + 1) / arg * jn - j; j = jn; jn = nx; }
        res = jn;
      }
      dst[l * NK + k] = g_norms[l * NK + k] * res;
    }
  }
}

__global__ void rbf_kernel(const float* __restrict__ dist, float* __restrict__ tab, int E) {
  int e = blockIdx.x * blockDim.x + threadIdx.x;
  if (e >= E) return;
  float* row = tab + (size_t)e * RBF_PAD;
  compute_rbf_row(dist[e] / 5.0f, row);
#pragma unroll
  for (int p = NRBF; p < RBF_PAD; ++p) row[p] = 0.0f;
}

// ---------------- async global->LDS gather (CDNA5) -------------------------
typedef int v4i_a __attribute__((ext_vector_type(4)));

__device__ inline void async_cp16(const float* g, float* l) {
#if __has_builtin(__builtin_amdgcn_global_load_async_to_lds_b128)
  // Signature (from hipcc diagnostic): param 0 is AS1 pointer to v4i.
  __builtin_amdgcn_global_load_async_to_lds_b128(
      (__attribute__((address_space(1))) v4i_a*)g,
      (__attribute__((address_space(3))) v4i_a*)l, 0, 0);
#else
  unsigned loff = (unsigned)(__SIZE_TYPE__)(__attribute__((address_space(3))) float*)l;
  asm volatile("global_load_async_to_lds_b128 %0, %1, off"
               :: "v"(loff), "v"(g) : "memory");
#endif
}
__device__ inline void wait_async0() {
#if __has_builtin(__builtin_amdgcn_s_wait_asynccnt)
  __builtin_amdgcn_s_wait_asynccnt(0);
#else
  asm volatile("s_wait_asynccnt 0" ::: "memory");
#endif
}

// ---------------- main kernel ----------------------------------------------
template <bool USE_TAB>
__global__ __launch_bounds__(128) void main_kernel(
    const float* __restrict__ angle, const float* __restrict__ phi,
    const int* __restrict__ idx_kj, const float* __restrict__ dist,
    const float* __restrict__ tab, float* __restrict__ out, int T) {
  __shared__ float s_cbf[4][32 * NCBF];     // stride 49 (odd) -> conflict-free writes
  __shared__ float s_rbf[4][32 * RBF_PAD];  // 192B rows, 16B aligned for async b128

  const int w = threadIdx.x >> 5;
  const int lane = threadIdx.x & 31;
  const int tBase = blockIdx.x * 128 + (w << 5);
  const int t = tBase + lane;
  float* cb = s_cbf[w];
  float* rb = s_rbf[w];

  if (t < T) {
    int idx = idx_kj[t];
    float* lrow = rb + lane * RBF_PAD;
    if (USE_TAB) {
      const float* grow = tab + (size_t)idx * RBF_PAD;
#pragma unroll
      for (int i = 0; i < RBF_PAD / 4; ++i) async_cp16(grow + 4 * i, lrow + 4 * i);
    } else {
      compute_rbf_row(dist[idx] / 5.0f, lrow);
    }

    // ---- cbf: real spherical harmonics (n=7), mirrors reference ops ----
    float th = angle[t], ph = phi[t];
    float s, z;
    sincosf(th, &s, &z);  // s = sin(theta), z = cos(theta)
    float P[NS][NS];
    P[0][0] = 1.0f;
    P[1][0] = z;
#pragma unroll
    for (int l = 2; l < NS; ++l)
      P[l][0] = ((float)(2 * l - 1) * z * P[l - 1][0] - (float)(l - 1) * P[l - 2][0]) / (float)l;
#pragma unroll
    for (int m = 1; m < NS; ++m) {
      P[m][m] = (1.0f - 2.0f * (float)m) * P[m - 1][m - 1];
      if (m + 1 < NS) P[m + 1][m] = (float)(2 * m + 1) * z * P[m][m];
#pragma unroll
      for (int l = m + 2; l < NS; ++l)
        P[l][m] = ((float)(2 * l - 1) * z * P[l - 1][m] - (float)(l + m - 1) * P[l - 2][m]) / (float)(l - m);
    }
    float cm[NS], sm[NS], spow[NS];
    spow[0] = 1.0f;
#pragma unroll
    for (int m = 1; m < NS; ++m) {
      sincosf((float)m * ph, &sm[m], &cm[m]);
      spow[m] = spow[m - 1] * s;
    }
    float* crow = cb + lane * NCBF;
    int c = 0;
    crow[c++] = 0.28209479177387814f;  // 0.5/sqrt(pi)
#pragma unroll
    for (int l = 1; l < NS; ++l) {
      crow[c++] = PREF.p0[l] * P[l][0];
#pragma unroll
      for (int m = 1; m <= l; ++m) crow[c++] = PREF.ps[l][m] * spow[m] * cm[m] * P[l][m];
#pragma unroll
      for (int m = l; m >= 1; --m) crow[c++] = PREF.ps[l][m] * spow[m] * sm[m] * P[l][m];
    }
  }

  if (USE_TAB) wait_async0();
  __syncthreads();

  // ---- transpose + coalesced float2 row stores ----
#pragma unroll 1
  for (int tt = 0; tt < 32; ++tt) {
    int trow = tBase + tt;
    if (trow >= T) break;  // uniform within the wave
    const float* cr = cb + tt * NCBF;
    const float* rr = rb + tt * RBF_PAD;
    float2* orow = (float2*)(out + (size_t)trow * OUT_ROW);
#pragma unroll
    for (int it = 0; it < 5; ++it) {
      int p = it * 32 + lane;  // pair index, 147 pairs per row
      if (p < 147) {
        int ab = p / 3;               // cbf index (0..48); each ab owns 3 pairs
        int r = (p - ab * 3) * 2;     // 0, 2, 4
        int b = ab % 7;
        float cv = cr[ab];
        float2 v;
        v.x = cv * rr[b * NK + r];
        v.y = cv * rr[b * NK + r + 1];
        orow[p] = v;
      }
    }
  }
}

// ---------------- launcher -------------------------------------------------
extern "C" void kernel_launch(void* const* d_in, const int* in_sizes, int n_in,
                              void* d_out, int out_size, void* d_ws, size_t ws_size,
                              hipStream_t stream) {
  const float* dist  = (const float*)d_in[0];
  const float* angle = (const float*)d_in[1];
  const float* phi   = (const float*)d_in[2];
  const int*   idx   = (const int*)d_in[3];
  float* out = (float*)d_out;
  const int E = in_sizes[0];
  const int T = in_sizes[1];

  zeros_kernel<<<1, 64, 0, stream>>>();

  const size_t tabBytes = (size_t)E * RBF_PAD * sizeof(float);
  const bool useTab = (d_ws != nullptr) && (ws_size >= tabBytes);
  if (useTab) {
    float* tab = (float*)d_ws;
    rbf_kernel<<<(E + 255) / 256, 256, 0, stream>>>(dist, tab, E);
    main_kernel<true><<<(T + 127) / 128, 128, 0, stream>>>(angle, phi, idx, dist, tab, out, T);
  } else {
    main_kernel<false><<<(T + 127) / 128, 128, 0, stream>>>(angle, phi, idx, dist, nullptr, out, T);
  }
}